// Huber_normalised_56822417326370
// MI455X (gfx1250) — compile-verified
//
#include <hip/hip_runtime.h>
#include <cstdint>

// Problem constants (from reference setup_inputs)
#define B_  16
#define C_  3
#define H_  512
#define W_  512
#define NPLANES        (B_ * C_)       // 48
#define PLANE_ELEMS    (H_ * W_)       // 262144
#define ROWS           8               // rows of output computed per block
#define SROWS          (ROWS + 1)      // staged rows incl. 1 halo row
#define BLKS_PER_PLANE (H_ / ROWS)     // 64
#define THREADS        256
#define VEC4_PER_TENSOR (SROWS * (W_ / 4))    // 9 * 128 = 1152
#define TOTAL_VEC4      (2 * VEC4_PER_TENSOR) // 2304 = 9 * 256
#define LOAD_ITERS      (TOTAL_VEC4 / THREADS) // 9, exact -> no EXEC masking

// huber(d) = m*(|d| - 0.5*m), m = min(|d|,1):
//   |d|<1 : m=|d| -> |d| - 0.5|d| is exact 0.5|d| -> 0.5 d^2   (matches ref)
//   |d|>=1: m=1   -> |d| - 0.5                                 (matches ref)
// Lowers to v_min_num_f32 (abs src mod) + 2 FMAs.
__device__ __forceinline__ float huber1(float d) {
    float a = fabsf(d);
    float m = fminf(a, 1.0f);
    return m * __builtin_fmaf(-0.5f, m, a);
}

__global__ __launch_bounds__(THREADS)
void grad_huber_partial(const float* __restrict__ in,
                        const float* __restrict__ tg,
                        float* __restrict__ part) {
    // 2 tensors x 9 rows x 512 floats = 36864 bytes of LDS
    __shared__ __align__(16) float smem[2 * SROWS * W_];

    const int tid   = threadIdx.x;
    const int plane = blockIdx.y;          // 0..47
    const int h0    = blockIdx.x * ROWS;   // first output row of this tile

    const uint64_t inb  = (uint64_t)(uintptr_t)in + (uint64_t)plane * (PLANE_ELEMS * 4u);
    const uint64_t tgb  = (uint64_t)(uintptr_t)tg + (uint64_t)plane * (PLANE_ELEMS * 4u);
    // addrspace(3) generic pointer: low 32 bits are the LDS byte offset
    const uint32_t lds0 = (uint32_t)(uintptr_t)(&smem[0]);

    // ---- Async stage: global -> LDS, one B128 (4 floats) per lane per iter ----
    // LDS row q holds global row clamp(h0 - 1 + q, 0); halo in row 0.
    #pragma unroll
    for (int i = 0; i < LOAD_ITERS; ++i) {
        int j    = i * THREADS + tid;                 // 0..2303
        int tsel = (j >= VEC4_PER_TENSOR) ? 1 : 0;    // 0 = input, 1 = target
        int k    = j - tsel * VEC4_PER_TENSOR;        // 0..1151
        int row  = k >> 7;                            // staged row 0..8
        int c4   = k & 127;                           // vec4 column
        int grow = h0 - 1 + row;
        if (grow < 0) grow = 0;                       // only for h0==0 halo (unused)
        uint64_t gaddr = (tsel ? tgb : inb) +
                         (uint64_t)(uint32_t)((grow * W_ + c4 * 4) * 4);
        uint32_t laddr = lds0 +
                         (uint32_t)(((tsel * SROWS + row) * W_ + c4 * 4) * 4);
        asm volatile("global_load_async_to_lds_b128 %0, %1, off"
                     :: "v"(laddr), "v"(gaddr) : "memory");
    }
    asm volatile("s_wait_asynccnt 0" ::: "memory");
    __syncthreads();

    const float* __restrict__ Lin = &smem[0];
    const float* __restrict__ Ltg = &smem[SROWS * W_];

    // ---- Compute: each thread owns a 16-pixel horizontal run ----
    float s1 = 0.0f, s2 = 0.0f;
    {
        const int r  = tid >> 5;            // row within tile, 0..7
        const int w0 = (tid & 31) << 4;     // segment start column, 16-aligned
        const int lr = r + 1;               // LDS row of global row h0+r
        const int lu = (h0 + r == 0) ? (lr + 1) : (lr - 1); // gy neighbor row
        const float* rowInC = Lin + lr * W_;
        const float* rowTgC = Ltg + lr * W_;
        const float* rowInU = Lin + lu * W_;
        const float* rowTgU = Ltg + lu * W_;

        // left neighbor of pixel w0 (reflection: w==0 uses x[1])
        const int wl = (w0 == 0) ? 1 : (w0 - 1);
        float prevTg = rowTgC[wl];
        float prevE  = rowInC[wl] - prevTg;   // e = in - tg of left neighbor

        #pragma unroll
        for (int c = 0; c < 4; ++c) {
            float4 iC = *reinterpret_cast<const float4*>(rowInC + w0 + 4 * c);
            float4 tC = *reinterpret_cast<const float4*>(rowTgC + w0 + 4 * c);
            float4 iU = *reinterpret_cast<const float4*>(rowInU + w0 + 4 * c);
            float4 tU = *reinterpret_cast<const float4*>(rowTgU + w0 + 4 * c);
            float ic[4] = {iC.x, iC.y, iC.z, iC.w};
            float tc[4] = {tC.x, tC.y, tC.z, tC.w};
            float iu[4] = {iU.x, iU.y, iU.z, iU.w};
            float tu[4] = {tU.x, tU.y, tU.z, tU.w};
            #pragma unroll
            for (int i = 0; i < 4; ++i) {
                float e   = ic[i] - tc[i];     // (in - tg) at center
                float eU  = iu[i] - tu[i];     // (in - tg) at up-neighbor
                float gxt = prevTg - tc[i];    // target gx
                float gyt = tu[i]  - tc[i];    // target gy
                float dgx = prevE  - e;        // grad(in)-grad(tg) == grad(in-tg)
                float dgy = eU     - e;
                s1 += huber1(dgx) + huber1(dgy);
                s2 += huber1(gxt) + huber1(gyt);
                prevE  = e;
                prevTg = tc[i];
            }
        }
    }

    // ---- Block reduction (wave32 shuffles, then cross-wave via LDS) ----
    for (int o = 16; o > 0; o >>= 1) {
        s1 += __shfl_down(s1, o, 32);
        s2 += __shfl_down(s2, o, 32);
    }
    __shared__ float red[2 * (THREADS / 32)];
    int wave = tid >> 5;
    if ((tid & 31) == 0) { red[wave] = s1; red[(THREADS / 32) + wave] = s2; }
    __syncthreads();
    if (tid == 0) {
        float a = 0.0f, b = 0.0f;
        #pragma unroll
        for (int i = 0; i < THREADS / 32; ++i) {
            a += red[i];
            b += red[(THREADS / 32) + i];
        }
        int slot = plane * BLKS_PER_PLANE + blockIdx.x;
        part[slot * 2 + 0] = a;
        part[slot * 2 + 1] = b;
    }
}

__global__ void grad_huber_finalize(const float* __restrict__ part,
                                    float* __restrict__ out) {
    __shared__ float ratio[NPLANES];
    int t = threadIdx.x;
    if (t < NPLANES) {
        float a = 0.0f, b = 0.0f;
        for (int i = 0; i < BLKS_PER_PLANE; ++i) {
            a += part[(t * BLKS_PER_PLANE + i) * 2 + 0];
            b += part[(t * BLKS_PER_PLANE + i) * 2 + 1];
        }
        // huber_mean/(gt_mean+1e-4) == sum1/(sum2 + 1e-4*N), N = 2*H*W
        ratio[t] = a / (b + 1e-4f * (float)(2 * H_ * W_));
    }
    __syncthreads();
    if (t == 0) {
        float s = 0.0f;
        for (int i = 0; i < NPLANES; ++i) s += ratio[i];
        out[0] = s / (float)NPLANES;
    }
}

extern "C" void kernel_launch(void* const* d_in, const int* in_sizes, int n_in,
                              void* d_out, int out_size, void* d_ws, size_t ws_size,
                              hipStream_t stream) {
    (void)in_sizes; (void)n_in; (void)out_size; (void)ws_size;
    const float* in = (const float*)d_in[0];
    const float* tg = (const float*)d_in[1];
    float* part = (float*)d_ws;  // 48*64*2 floats = 24576 bytes, fully overwritten each call

    dim3 grid(BLKS_PER_PLANE, NPLANES);   // (64, 48) = 3072 blocks
    grad_huber_partial<<<grid, THREADS, 0, stream>>>(in, tg, part);
    grad_huber_finalize<<<1, 64, 0, stream>>>(part, (float*)d_out);
}